// VQEmbedding_781684048211
// MI455X (gfx1250) — compile-verified
//
#include <hip/hip_runtime.h>

typedef __attribute__((ext_vector_type(2))) float v2f;
typedef __attribute__((ext_vector_type(8))) float v8f;
typedef int v4i __attribute__((__vector_size__(16)));   // matches builtin param pointee

#define D_DIM   256
#define K_CODES 8192
#define N_ROWS  32768
#define CHUNK   16            // codebook rows per staged chunk
#define NCHUNKS (K_CODES / CHUNK)
#define LDS_STRIDE 260        // padded row stride (floats): 4*r mod 64 distinct -> conflict free
#define ROWS_PER_BLOCK 128    // 8 waves * 16 rows

#define AS1 __attribute__((address_space(1)))
#define AS3 __attribute__((address_space(3)))

#if __has_builtin(__builtin_amdgcn_global_load_async_to_lds_b128) && \
    __has_builtin(__builtin_amdgcn_s_wait_asynccnt)
#define USE_ASYNC 1
#else
#define USE_ASYNC 0
#endif

// ---------------------------------------------------------------------------
// Kernel 1: w_sq[k] = sum_d codebook[k][d]^2 ; also zero the loss accumulator.
// ---------------------------------------------------------------------------
__global__ __launch_bounds__(256) void wsq_kernel(const float* __restrict__ cb,
                                                  float* __restrict__ wsq,
                                                  float* __restrict__ accum) {
    if (blockIdx.x == 0 && threadIdx.x == 0) accum[0] = 0.0f;
    const int wave = threadIdx.x >> 5;
    const int lane = threadIdx.x & 31;
    const int row  = blockIdx.x * 8 + wave;          // grid = K/8 blocks
    const float4* p = (const float4*)(cb + (size_t)row * D_DIM);
    float4 x = p[lane];
    float4 y = p[lane + 32];
    float s = x.x*x.x + x.y*x.y + x.z*x.z + x.w*x.w
            + y.x*y.x + y.y*y.y + y.z*y.z + y.w*y.w;
#pragma unroll
    for (int m = 16; m >= 1; m >>= 1) s += __shfl_xor(s, m, 32);
    if (lane == 0) wsq[row] = s;
}

// ---------------------------------------------------------------------------
// Kernel 2: argmin over codes of (w_sq[k] - 2 * <h_row, codebook_k>) via
// V_WMMA_F32_16X16X4_F32. A-tile (16 rows x 256) lives in registers in the
// ISA A-layout. Codebook chunks are double-buffered in LDS via the async
// Global->LDS DMA path (ASYNCcnt) when available; otherwise a register-
// prefetch software pipeline hides the global latency.
// ---------------------------------------------------------------------------
__global__ __launch_bounds__(256) void argmin_kernel(const float* __restrict__ h,
                                                     const float* __restrict__ cb,
                                                     const float* __restrict__ wsq,
                                                     int* __restrict__ indices) {
    __shared__ float lds[2 * CHUNK * LDS_STRIDE];    // 33,280 bytes (two buffers)

    const int tid    = threadIdx.x;
    const int wave   = tid >> 5;
    const int lane   = tid & 31;
    const int lane16 = lane & 15;
    const int half   = lane >> 4;                    // 0: K=0,1 / M=j ; 1: K=2,3 / M=j+8
    const int rowBase = blockIdx.x * ROWS_PER_BLOCK + wave * 16;

#if USE_ASYNC
    // Issue async DMA for chunk 0 into buffer 0 before touching A, so the
    // first chunk streams in while we fill the A registers.
    {
#pragma unroll
        for (int k = 0; k < 4; ++k) {
            const int i = tid + k * 256;
            const int r = i >> 6, f = i & 63;
            const float* gsrc = cb + (size_t)r * D_DIM + f * 4;
            const float* ldst = &lds[r * LDS_STRIDE + f * 4];
            __builtin_amdgcn_global_load_async_to_lds_b128(
                (AS1 v4i*)(uintptr_t)gsrc,
                (AS3 v4i*)(uint32_t)(uintptr_t)ldst, 0, 0);
        }
    }
#endif

    // ---- A tile into registers: lane holds h[rowBase+lane16][d*4 + half*2 + {0,1}]
    v2f a[64];
    const float* arow = h + (size_t)(rowBase + lane16) * D_DIM + half * 2;
#pragma unroll
    for (int d = 0; d < 64; ++d)
        a[d] = *(const v2f*)(arow + d * 4);

    float minV[8];
    int   minI[8];
#pragma unroll
    for (int j = 0; j < 8; ++j) { minV[j] = __builtin_inff(); minI[j] = 0; }

    // 16x16 WMMA tile against one staged 16-code chunk.
    auto compute_chunk = [&](int c) {
        const float* buf  = lds + (c & 1) * (CHUNK * LDS_STRIDE);
        const float* brow = buf + lane16 * LDS_STRIDE + half * 2;
        v8f acc = {};
#pragma unroll
        for (int d = 0; d < 64; ++d) {
            v2f b = *(const v2f*)(brow + d * 4);
            acc = __builtin_amdgcn_wmma_f32_16x16x4_f32(
                      false, a[d], false, b, (short)0, acc, false, false);
        }
        const int code = c * CHUNK + lane16;         // C-layout column N = lane16
        const float w  = wsq[code];
#pragma unroll
        for (int j = 0; j < 8; ++j) {                // row M = j + 8*half
            const float dist = w - 2.0f * acc[j];
            if (dist < minV[j]) { minV[j] = dist; minI[j] = code; }
        }
    };

#if USE_ASYNC
    // Double-buffered pipeline: issue chunk c+1, wait for chunk c (in-order
    // completion => asynccnt <= 4 keeps the new 4 in flight), then compute.
    for (int c = 0; c < NCHUNKS - 1; ++c) {
        const int buf = (c + 1) & 1;
#pragma unroll
        for (int k = 0; k < 4; ++k) {
            const int i = tid + k * 256;
            const int r = i >> 6, f = i & 63;
            const float* gsrc = cb + (size_t)((c + 1) * CHUNK + r) * D_DIM + f * 4;
            const float* ldst = &lds[buf * (CHUNK * LDS_STRIDE) + r * LDS_STRIDE + f * 4];
            __builtin_amdgcn_global_load_async_to_lds_b128(
                (AS1 v4i*)(uintptr_t)gsrc,
                (AS3 v4i*)(uint32_t)(uintptr_t)ldst, 0, 0);
        }
        __builtin_amdgcn_s_wait_asynccnt(4);
        __syncthreads();                 // chunk c visible to all waves
        compute_chunk(c);
        __syncthreads();                 // buffer c&1 free for chunk c+2
    }
    __builtin_amdgcn_s_wait_asynccnt(0);
    __syncthreads();
    compute_chunk(NCHUNKS - 1);
#else
    // Fallback: register-prefetch software pipeline (plain loads).
    float4 pre[4];
#pragma unroll
    for (int k = 0; k < 4; ++k) {
        const int i = tid + k * 256;
        const int r = i >> 6, f = i & 63;
        pre[k] = *(const float4*)(cb + (size_t)r * D_DIM + f * 4);
    }
    for (int c = 0; c < NCHUNKS; ++c) {
        __syncthreads();                 // buffer free
#pragma unroll
        for (int k = 0; k < 4; ++k) {
            const int i = tid + k * 256;
            const int r = i >> 6, f = i & 63;
            *(float4*)&lds[r * LDS_STRIDE + f * 4] = pre[k];
        }
        if (c + 1 < NCHUNKS) {
#pragma unroll
            for (int k = 0; k < 4; ++k) {
                const int i = tid + k * 256;
                const int r = i >> 6, f = i & 63;
                pre[k] = *(const float4*)(cb + (size_t)((c + 1) * CHUNK + r) * D_DIM + f * 4);
            }
        }
        __syncthreads();
        {
            const float* brow = lds + lane16 * LDS_STRIDE + half * 2;
            v8f acc = {};
#pragma unroll
            for (int d = 0; d < 64; ++d) {
                v2f b = *(const v2f*)(brow + d * 4);
                acc = __builtin_amdgcn_wmma_f32_16x16x4_f32(
                          false, a[d], false, b, (short)0, acc, false, false);
            }
            const int code = c * CHUNK + lane16;
            const float w  = wsq[code];
#pragma unroll
            for (int j = 0; j < 8; ++j) {
                const float dist = w - 2.0f * acc[j];
                if (dist < minV[j]) { minV[j] = dist; minI[j] = code; }
            }
        }
    }
#endif

    // reduce (min, idx) across the 16 lanes of each half; tie-break lowest index
#pragma unroll
    for (int j = 0; j < 8; ++j) {
        float v = minV[j];
        int   ix = minI[j];
#pragma unroll
        for (int m = 8; m >= 1; m >>= 1) {
            const float ov = __shfl_xor(v, m, 32);
            const int   oi = __shfl_xor(ix, m, 32);
            if (ov < v || (ov == v && oi < ix)) { v = ov; ix = oi; }
        }
        if (lane16 == 0) indices[rowBase + half * 8 + j] = ix;
    }
}

// ---------------------------------------------------------------------------
// Kernel 3: gather quantized rows, write output, accumulate sum((h-q)^2).
// ---------------------------------------------------------------------------
__global__ __launch_bounds__(256) void gather_loss_kernel(const float* __restrict__ h,
                                                          const float* __restrict__ cb,
                                                          const int* __restrict__ indices,
                                                          float* __restrict__ out,
                                                          float* __restrict__ accum) {
    __shared__ float wsum[8];
    const int wave = threadIdx.x >> 5;
    const int lane = threadIdx.x & 31;
    const int row  = blockIdx.x * 8 + wave;          // grid = N/8 blocks
    const int idx  = indices[row];
    const float4* q  = (const float4*)(cb + (size_t)idx * D_DIM);
    const float4* hp = (const float4*)(h + (size_t)row * D_DIM);
    float4*       op = (float4*)(out + (size_t)row * D_DIM);
    float s = 0.0f;
#pragma unroll
    for (int t = 0; t < 2; ++t) {
        const int e = lane + t * 32;
        const float4 qa = q[e];
        const float4 ha = hp[e];
        op[e] = qa;
        const float dx = ha.x - qa.x, dy = ha.y - qa.y;
        const float dz = ha.z - qa.z, dw = ha.w - qa.w;
        s += dx*dx + dy*dy + dz*dz + dw*dw;
    }
#pragma unroll
    for (int m = 16; m >= 1; m >>= 1) s += __shfl_xor(s, m, 32);
    if (lane == 0) wsum[wave] = s;
    __syncthreads();
    if (threadIdx.x == 0) {
        float t = 0.0f;
#pragma unroll
        for (int i = 0; i < 8; ++i) t += wsum[i];
        atomicAdd(accum, t);
    }
}

// ---------------------------------------------------------------------------
// Kernel 4: finalize the two scalar losses.
// ---------------------------------------------------------------------------
__global__ void finalize_kernel(const float* __restrict__ accum, float* __restrict__ out) {
    const float mse = accum[0] / (float)((size_t)N_ROWS * D_DIM);
    out[(size_t)N_ROWS * D_DIM + 0] = 0.25f * mse;   // commitment_loss
    out[(size_t)N_ROWS * D_DIM + 1] = mse;           // codebook_loss
}

extern "C" void kernel_launch(void* const* d_in, const int* in_sizes, int n_in,
                              void* d_out, int out_size, void* d_ws, size_t ws_size,
                              hipStream_t stream) {
    const float* h  = (const float*)d_in[0];   // [32,1024,256] f32
    const float* cb = (const float*)d_in[1];   // [8192,256]    f32
    float* out = (float*)d_out;                // [N*D] quantized + 2 scalars

    // workspace layout
    float* wsq     = (float*)d_ws;             // 8192 floats
    int*   indices = (int*)(wsq + K_CODES);    // 32768 ints
    float* accum   = (float*)(indices + N_ROWS);

    wsq_kernel<<<K_CODES / 8, 256, 0, stream>>>(cb, wsq, accum);
    argmin_kernel<<<N_ROWS / ROWS_PER_BLOCK, 256, 0, stream>>>(h, cb, wsq, indices);
    gather_loss_kernel<<<N_ROWS / 8, 256, 0, stream>>>(h, cb, indices, out, accum);
    finalize_kernel<<<1, 1, 0, stream>>>(accum, out);
}